// GINConv2d_7138235646509
// MI455X (gfx1250) — compile-verified
//
#include <hip/hip_runtime.h>

typedef float v2f __attribute__((ext_vector_type(2)));
typedef float v8f __attribute__((ext_vector_type(8)));

#define BB 8
#define CC 256
#define NN 4096
#define KK 16
#define NT 64   // nodes (columns) per block tile

// Fused GIN gather + (1+eps)x + neighbor-sum, then W @ h + b with ReLU.
// LDS holds the h tile in WMMA-B-fragment order (channel-paired):
//   hp[(c/2)*(2*NT) + j*2 + (c&1)]
// so each B fragment is one ds_load_b64 into an even-aligned VGPR pair.
__global__ __launch_bounds__(256) void gin_fused_kernel(
    const float* __restrict__ x,        // [B, C, N]
    const long long* __restrict__ edge, // [2, B, N, K]; plane 0 used
    const float* __restrict__ eps,      // [1]
    const float* __restrict__ Wm,       // [C, C] row-major
    const float* __restrict__ bias,     // [C]
    float* __restrict__ out)            // [B, C, N]
{
    __shared__ float hp[CC * NT];       // 64 KB h tile, channel-paired layout

    const int tid = threadIdx.x;
    const int blk = blockIdx.x;                 // b * (N/NT) + ntile
    const int b   = blk / (NN / NT);
    const int n0  = (blk % (NN / NT)) * NT;

    const float scale = 1.0f + eps[0];
    const float* xb = x + (size_t)b * CC * NN;

    // ---------------- Phase 1: gather + GIN update into LDS ----------------
    {
        const int j   = tid & (NT - 1);         // node within tile (coalesced n)
        const int cch = tid >> 6;               // 0..3 channel chunk
        const int n   = n0 + j;

        int idx[KK];
        const long long* e = edge + ((size_t)b * NN + n) * KK;  // edge_index[0]
        #pragma unroll
        for (int k = 0; k < KK; ++k) idx[k] = (int)e[k];

        const int cbeg = cch * (CC / 4);
        for (int c = cbeg; c < cbeg + (CC / 4); c += 2) {
            const float* r0 = xb + (size_t)c * NN;
            const float* r1 = r0 + NN;
            float a0 = scale * r0[n];           // coalesced along n
            float a1 = scale * r1[n];
            #pragma unroll
            for (int k = 0; k < KK; ++k) {      // random, L2-hot (4MB/batch slice)
                a0 += r0[idx[k]];
                a1 += r1[idx[k]];
            }
            v2f pr; pr.x = a0; pr.y = a1;
            // paired store: lanes write consecutive b64 -> conflict-free
            *(v2f*)&hp[(c >> 1) * (2 * NT) + j * 2] = pr;
        }
    }
    __syncthreads();

    // ---------------- Phase 2: WMMA GEMM: relu(W @ h + bias) ----------------
    const int wave = tid >> 5;
    const int lane = tid & 31;
    const int half = lane >> 4;                 // 0 or 1 (lanes 16-31)
    const int l16  = lane & 15;
    float* ob = out + (size_t)b * CC * NN;

    for (int mi = 0; mi < 2; ++mi) {
        const int m0 = (wave * 2 + mi) * 16;    // 8 waves x 2 = 16 M-tiles

        // A fragment: W[m0 + l16][k + 2*half + v], v in {0,1} -> global b64
        const float* wrow = Wm + (m0 + l16) * CC + 2 * half;
        // B fragment base: pair row p = k/2 + half, lane column l16
        const float* hb0 = &hp[half * (2 * NT) + l16 * 2];

        v8f acc0 = {}, acc1 = {}, acc2 = {}, acc3 = {};

        for (int k = 0; k < CC; k += 4) {
            v2f a;
            a.x = wrow[k];
            a.y = wrow[k + 1];

            const float* hk = hb0 + (k >> 1) * (2 * NT);
            v2f b0 = *(const v2f*)(hk + 0);     // ntile 0: one ds_load_b64 each
            v2f b1 = *(const v2f*)(hk + 32);    // ntile 1 (+128 B)
            v2f b2 = *(const v2f*)(hk + 64);    // ntile 2 (+256 B)
            v2f b3 = *(const v2f*)(hk + 96);    // ntile 3 (+384 B)

            acc0 = __builtin_amdgcn_wmma_f32_16x16x4_f32(false, a, false, b0, (short)0, acc0, false, false);
            acc1 = __builtin_amdgcn_wmma_f32_16x16x4_f32(false, a, false, b1, (short)0, acc1, false, false);
            acc2 = __builtin_amdgcn_wmma_f32_16x16x4_f32(false, a, false, b2, (short)0, acc2, false, false);
            acc3 = __builtin_amdgcn_wmma_f32_16x16x4_f32(false, a, false, b3, (short)0, acc3, false, false);
        }

        // Epilogue: C/D layout -> M = r + 8*half, N = l16; add bias + ReLU
        #pragma unroll
        for (int r = 0; r < 8; ++r) {
            const int m = m0 + r + 8 * half;
            const float bv = bias[m];
            float* orow = ob + (size_t)m * NN + n0 + l16;
            orow[0]  = fmaxf(acc0[r] + bv, 0.0f);
            orow[16] = fmaxf(acc1[r] + bv, 0.0f);
            orow[32] = fmaxf(acc2[r] + bv, 0.0f);
            orow[48] = fmaxf(acc3[r] + bv, 0.0f);
        }
    }
}

extern "C" void kernel_launch(void* const* d_in, const int* in_sizes, int n_in,
                              void* d_out, int out_size, void* d_ws, size_t ws_size,
                              hipStream_t stream) {
    (void)in_sizes; (void)n_in; (void)d_ws; (void)ws_size; (void)out_size;
    const float*     x    = (const float*)d_in[0];
    const long long* edge = (const long long*)d_in[1];   // int64 edge_index
    const float*     eps  = (const float*)d_in[2];
    const float*     Wm   = (const float*)d_in[3];
    const float*     bias = (const float*)d_in[4];
    float*           out  = (float*)d_out;

    dim3 grid(BB * (NN / NT));   // 8 * 64 = 512 blocks
    dim3 block(256);             // 8 wave32 waves
    gin_fused_kernel<<<grid, block, 0, stream>>>(x, edge, eps, Wm, bias, out);
}